// stn_transformer_37203006718317
// MI455X (gfx1250) — compile-verified
//
#include <hip/hip_runtime.h>

typedef float v4f __attribute__((ext_vector_type(4)));

// STN bilinear sampler: x[32,256,256,32] f32 NHWC, theta[32,6] -> out[32,256,256,32]
//
// Memory-bound gather (AI ~1 FLOP/byte); each neighbor pixel is exactly one
// 128B cacheline, so lane/channel partitioning doesn't change traffic — only
// VALU redundancy. 2 threads per output pixel (16 channels = 64B each):
//  - weight/affine math computed 2x per pixel instead of 8x (3x less VALU/byte)
//  - 16 independent global_load_b128 per thread (deep MLP for latency hiding)
//  - theta reads are block-uniform -> scalar (s_load) path
//  - output streamed with non-temporal b128 stores to keep L2 for input reuse
__global__ __launch_bounds__(256) void stn_bilinear_kernel(
    const float* __restrict__ x,
    const float* __restrict__ theta,
    float* __restrict__ out) {
  constexpr int W = 256;   // input width
  constexpr int H = 256;   // input height
  constexpr int C = 32;    // channels
  constexpr int OW = 256;  // output width

  // 512 blocks per batch image (65536 pixels / 128 pixels-per-block).
  const int b     = blockIdx.x >> 9;
  const int pbase = (blockIdx.x & 511) << 7;     // 128 pixels per block
  const int p     = pbase + (threadIdx.x >> 1);  // pixel index within image
  const int h     = threadIdx.x & 1;             // half: channels 16h .. 16h+15

  const float t0 = theta[b * 6 + 0];
  const float t1 = theta[b * 6 + 1];
  const float t2 = theta[b * 6 + 2];
  const float t3 = theta[b * 6 + 3];
  const float t4 = theta[b * 6 + 4];
  const float t5 = theta[b * 6 + 5];

  const int oy = p >> 8;
  const int ox = p & (OW - 1);

  // linspace(-1, 1, 256): step = 2/255
  const float gx = -1.0f + (float)ox * (2.0f / 255.0f);
  const float gy = -1.0f + (float)oy * (2.0f / 255.0f);

  // Affine transform (theta reshaped [2,3]): row0 -> x, row1 -> y
  const float sxs = t0 * gx + t1 * gy + t2;
  const float sys = t3 * gx + t4 * gy + t5;

  // Map to pixel space
  const float sx = 0.5f * (sxs + 1.0f) * (float)W;
  const float sy = 0.5f * (sys + 1.0f) * (float)H;

  // Truncation toward zero, matching astype(int32)
  const int x0 = (int)sx;
  const int y0 = (int)sy;
  const int x1 = x0 + 1;
  const int y1 = y0 + 1;

  const int x0c = min(max(x0, 0), W - 1);
  const int x1c = min(max(x1, 0), W - 1);
  const int y0c = min(max(y0, 0), H - 1);
  const int y1c = min(max(y1, 0), H - 1);

  const float x0f = (float)x0c;
  const float x1f = (float)x1c;
  const float y0f = (float)y0c;
  const float y1f = (float)y1c;

  // Bilinear weights from *clamped* coords (matches reference)
  const float wa = (x1f - sx) * (y1f - sy);
  const float wb = (x1f - sx) * (sy - y0f);
  const float wc = (sx - x0f) * (y1f - sy);
  const float wd = (sx - x0f) * (sy - y0f);

  // Flat source pixel indices within the whole tensor
  const int rb = b << 16;  // b * H * W
  const int ia = rb + (y0c << 8) + x0c;
  const int ib = rb + (y1c << 8) + x0c;
  const int ic = rb + (y0c << 8) + x1c;
  const int id = rb + (y1c << 8) + x1c;

  const int co = h << 4;  // channel offset within pixel: 0 or 16 floats

  const size_t offA = (size_t)ia * C + co;
  const size_t offB = (size_t)ib * C + co;
  const size_t offC = (size_t)ic * C + co;
  const size_t offD = (size_t)id * C + co;
  const size_t dst  = ((size_t)(rb + p)) * C + co;

  // 16 channels = 4 float4 chunks; issue all 16 loads, blend, stream out.
  v4f pa[4], pb[4], pc[4], pd[4];
#pragma unroll
  for (int k = 0; k < 4; ++k) {
    pa[k] = *(const v4f*)(x + offA + 4 * k);
    pb[k] = *(const v4f*)(x + offB + 4 * k);
    pc[k] = *(const v4f*)(x + offC + 4 * k);
    pd[k] = *(const v4f*)(x + offD + 4 * k);
  }
#pragma unroll
  for (int k = 0; k < 4; ++k) {
    v4f acc = wa * pa[k];
    acc += wb * pb[k];
    acc += wc * pc[k];
    acc += wd * pd[k];
    __builtin_nontemporal_store(acc, (v4f*)(out + dst + 4 * k));
  }
}

extern "C" void kernel_launch(void* const* d_in, const int* in_sizes, int n_in,
                              void* d_out, int out_size, void* d_ws, size_t ws_size,
                              hipStream_t stream) {
  const float* x     = (const float*)d_in[0];
  const float* theta = (const float*)d_in[1];
  float* out         = (float*)d_out;

  // 32 images * 512 blocks/image, 256 threads (8 wave32) per block.
  dim3 grid(32 * 512);
  dim3 block(256);
  stn_bilinear_kernel<<<grid, block, 0, stream>>>(x, theta, out);
}